// CrossModalIsomorphismBridge_56126632624519
// MI455X (gfx1250) — compile-verified
//
#include <hip/hip_runtime.h>
#include <hip/hip_bf16.h>

#define NB 262144
#define OFF_EV   (NB * 128)
#define OFF_ET   (NB * 129)
#define OFF_EB   (NB * 130)
#define OFF_ISO  (NB * 131)
#define OFF_LC   (NB * 131 + 1)
#define OFF_DEC  (NB * 131 + 2)

typedef __attribute__((ext_vector_type(2))) float v2f;
typedef __attribute__((ext_vector_type(8))) float v8f;

#if defined(__has_builtin)
#if __has_builtin(__builtin_amdgcn_wmma_f32_16x16x4_f32)
#define HAVE_WMMA_F32X4 1
#endif
#endif

__device__ __forceinline__ float dot4(const float a[4], const float b[4]) {
  return a[0]*b[0] + a[1]*b[1] + a[2]*b[2] + a[3]*b[3];
}

__device__ __forceinline__ void norm4(float q[4]) {
  float inv = 1.0f / (sqrtf(dot4(q, q)) + 1e-8f);
  q[0] *= inv; q[1] *= inv; q[2] *= inv; q[3] *= inv;
}

__device__ __forceinline__ float det4(float m[4][4]) {
  float s0 = m[0][0]*m[1][1] - m[0][1]*m[1][0];
  float s1 = m[0][0]*m[1][2] - m[0][2]*m[1][0];
  float s2 = m[0][0]*m[1][3] - m[0][3]*m[1][0];
  float s3 = m[0][1]*m[1][2] - m[0][2]*m[1][1];
  float s4 = m[0][1]*m[1][3] - m[0][3]*m[1][1];
  float s5 = m[0][2]*m[1][3] - m[0][3]*m[1][2];
  float c5 = m[2][2]*m[3][3] - m[2][3]*m[3][2];
  float c4 = m[2][1]*m[3][3] - m[2][3]*m[3][1];
  float c3 = m[2][1]*m[3][2] - m[2][2]*m[3][1];
  float c2 = m[2][0]*m[3][3] - m[2][3]*m[3][0];
  float c1 = m[2][0]*m[3][2] - m[2][2]*m[3][0];
  float c0 = m[2][0]*m[3][1] - m[2][1]*m[3][0];
  return s0*c5 - s1*c4 + s2*c3 + s3*c2 - s4*c1 + s5*c0;
}

// One wave32 per row; lane = quaternion index (32 quats * 4 comps = 128).
__global__ __launch_bounds__(256) void bridge_main(
    const float* __restrict__ vision, const int* __restrict__ text,
    const float* __restrict__ Wv, const float* __restrict__ bv,
    const float* __restrict__ Wt, const float* __restrict__ bt,
    float* __restrict__ out, float* __restrict__ accum) {
  __shared__ float smat[8][3][4][4];
  __shared__ float sred[3][8];

  const int lane = threadIdx.x & 31;
  const int wib  = threadIdx.x >> 5;
  const int row  = blockIdx.x * 8 + wib;

  const float v0 = vision[row * 3 + 0];
  const float v1 = vision[row * 3 + 1];
  const float v2 = vision[row * 3 + 2];
  const float tv = (float)text[row];

  float vq[4], tq[4];
#pragma unroll
  for (int j = 0; j < 4; ++j) {
    int m = lane * 4 + j;
    vq[j] = fmaf(v0, Wv[m*3+0], fmaf(v1, Wv[m*3+1], fmaf(v2, Wv[m*3+2], bv[m])));
    tq[j] = fmaf(tv, Wt[m], bt[m]);
  }
  norm4(vq);
  norm4(tq);

  float bq[4];
#pragma unroll
  for (int j = 0; j < 4; ++j) bq[j] = 0.5f * (vq[j] + tq[j]);
  norm4(bq);

#pragma unroll
  for (int it = 0; it < 5; ++it) {
    float dv = dot4(vq, bq);
    float dt = dot4(tq, bq);
    dv = fminf(fmaxf(dv, -1.0f + 1e-6f), 1.0f - 1e-6f);
    dt = fminf(fmaxf(dt, -1.0f + 1e-6f), 1.0f - 1e-6f);
    float thv = acosf(dv), tht = acosf(dt);
    float pv[4], pt[4];
#pragma unroll
    for (int j = 0; j < 4; ++j) {
      pv[j] = vq[j] - dv * bq[j];
      pt[j] = tq[j] - dt * bq[j];
    }
    float sv = thv / (sqrtf(dot4(pv, pv)) + 1e-8f);
    float st = tht / (sqrtf(dot4(pt, pt)) + 1e-8f);
    float tg[4];
#pragma unroll
    for (int j = 0; j < 4; ++j) tg[j] = 0.5f * (sv * pv[j] + st * pt[j]);
    float tn = sqrtf(dot4(tg, tg));
    float ct = cosf(tn);
    float sn = sinf(tn) / (tn + 1e-8f);
#pragma unroll
    for (int j = 0; j < 4; ++j) bq[j] = ct * bq[j] + sn * tg[j];
    norm4(bq);
  }

  // Coalesced barycenter store: 32 lanes x float4 = 512B contiguous per wave.
  float4 bw; bw.x = bq[0]; bw.y = bq[1]; bw.z = bq[2]; bw.w = bq[3];
  *reinterpret_cast<float4*>(out + (size_t)row * 128 + lane * 4) = bw;

  // Quats 0,8,16,24 feed the 4x4 determinant matrices.
  if ((lane & 7) == 0) {
    int r = lane >> 3;
#pragma unroll
    for (int j = 0; j < 4; ++j) {
      smat[wib][0][r][j] = vq[j];
      smat[wib][1][r][j] = tq[j];
      smat[wib][2][r][j] = bq[j];
    }
  }
  __syncthreads();
  if (threadIdx.x < 8) {
    int w = threadIdx.x;
    int rrow = blockIdx.x * 8 + w;
    float ev = det4(smat[w][0]) < 0.0f ? 1.0f : 0.0f;
    float et = det4(smat[w][1]) < 0.0f ? 1.0f : 0.0f;
    float eb = det4(smat[w][2]) < 0.0f ? 1.0f : 0.0f;
    out[OFF_EV + rrow] = ev;
    out[OFF_ET + rrow] = et;
    out[OFF_EB + rrow] = eb;
    float d0 = ev - et, d1 = ev - eb, d2 = et - eb;
    sred[0][w] = fabsf(d0);
    sred[1][w] = d1 * d1;
    sred[2][w] = d2 * d2;
  }
  __syncthreads();
  if (threadIdx.x == 0) {
    float s0 = 0.f, s1 = 0.f, s2 = 0.f;
#pragma unroll
    for (int w = 0; w < 8; ++w) { s0 += sred[0][w]; s1 += sred[1][w]; s2 += sred[2][w]; }
    // All addends in {0,1}, totals < 2^24: float atomic sum is exact & order-independent.
    atomicAdd(&accum[0], s0);
    atomicAdd(&accum[1], s1);
    atomicAdd(&accum[2], s2);
  }
}

// decision = barycenter(16 rows x 128) @ Wd^T (128 x 2, padded to 16 cols),
// via 32 chained V_WMMA_F32_16X16X4_F32 per wave. EXEC all-1s around WMMAs.
__global__ __launch_bounds__(256) void bridge_decision(
    const float* __restrict__ bary, const float* __restrict__ Wd,
    const float* __restrict__ bd, float* __restrict__ dec) {
  const int lane = threadIdx.x & 31;
  const int wave = blockIdx.x * 8 + (threadIdx.x >> 5);
  const int rowBase = wave * 16;
  const int n = lane & 15;       // M for A / N for B,D
  const int khalf = lane >> 4;   // hi lanes hold K=2,3 of each tile
  const int mrow = rowBase + n;
#if HAVE_WMMA_F32X4
  v8f acc = {};
  const float mask = (n < 2) ? 1.0f : 0.0f;   // pad N=2..15 with zero columns
  const int wrow = (n < 2) ? n : 0;           // clamp address, zero via mask
#pragma unroll 4
  for (int kt = 0; kt < 32; ++kt) {
    const int k0 = kt * 4 + 2 * khalf;
    const float2 a2 = *reinterpret_cast<const float2*>(bary + (size_t)mrow * 128 + k0);
    const float2 w2 = *reinterpret_cast<const float2*>(Wd + wrow * 128 + k0);
    v2f a;  a[0] = a2.x;        a[1] = a2.y;
    v2f bm; bm[0] = w2.x * mask; bm[1] = w2.y * mask;
    acc = __builtin_amdgcn_wmma_f32_16x16x4_f32(
        false, a, false, bm, (short)0, acc, false, false);
  }
  if (n < 2) {  // lane n holds column n; VGPR r = row r (+8 for hi lanes)
    const float bias = bd[n];
#pragma unroll
    for (int r = 0; r < 8; ++r) {
      int rr = rowBase + khalf * 8 + r;
      dec[rr * 2 + n] = acc[r] + bias;
    }
  }
#else
  int rrow = rowBase + (lane >> 1);
  int col = lane & 1;
  float s = bd[col];
  for (int k = 0; k < 128; ++k) s += bary[(size_t)rrow * 128 + k] * Wd[col * 128 + k];
  dec[rrow * 2 + col] = s;
#endif
}

__global__ void bridge_final(const float* __restrict__ accum, float* __restrict__ out) {
  out[OFF_ISO] = sqrtf(accum[1]) + sqrtf(accum[2]);
  out[OFF_LC]  = accum[0] / (float)NB;
}

extern "C" void kernel_launch(void* const* d_in, const int* in_sizes, int n_in,
                              void* d_out, int out_size, void* d_ws, size_t ws_size,
                              hipStream_t stream) {
  const float* vision = (const float*)d_in[0];
  const int*   text   = (const int*)d_in[1];
  const float* Wv     = (const float*)d_in[2];
  const float* bv     = (const float*)d_in[3];
  const float* Wt     = (const float*)d_in[4];
  const float* bt     = (const float*)d_in[5];
  const float* Wd     = (const float*)d_in[6];
  const float* bd     = (const float*)d_in[7];
  float* out   = (float*)d_out;
  float* accum = (float*)d_ws;

  hipMemsetAsync(accum, 0, 3 * sizeof(float), stream);  // capture-safe memset node
  bridge_main<<<NB / 8, 256, 0, stream>>>(vision, text, Wv, bv, Wt, bt, out, accum);
  bridge_decision<<<NB / 128, 256, 0, stream>>>(out, Wd, bd, out + OFF_DEC);
  bridge_final<<<1, 1, 0, stream>>>(accum, out);
}